// EquivMPRepresentation_77120432767618
// MI455X (gfx1250) — compile-verified
//
#include <hip/hip_runtime.h>
#include <hip/hip_bf16.h>
#include <math.h>

#define N_NODES 10000
#define N_EDGES 320000
#define FEAT 32
#define NBASIS 8
#define RMAX 6.0f

typedef __attribute__((ext_vector_type(2))) float v2f;
typedef __attribute__((ext_vector_type(8))) float v8f;
typedef __attribute__((ext_vector_type(4))) int v4i;

#if defined(__has_builtin)
#if __has_builtin(__builtin_amdgcn_global_load_async_to_lds_b128) && \
    __has_builtin(__builtin_amdgcn_s_wait_asynccnt)
#define USE_ASYNC_LDS 1
typedef __attribute__((address_space(1))) v4i gv4i;  // global int4
typedef __attribute__((address_space(3))) v4i lv4i;  // LDS int4
#endif
#endif

// ---------------------------------------------------------------------------
// Path tables: (l1,l2,l3) per iteration, matching the reference PATHS0/1/2.
// ---------------------------------------------------------------------------
__device__ __constant__ signed char PATHS_L[21][3] = {
    // it0 (3 paths)
    {0,0,0},{0,1,1},{0,2,2},
    // it1 (15 paths, l1-major / l2 / l3 order like the python comprehension)
    {0,0,0},{0,1,1},{0,2,2},
    {1,0,1},{1,1,0},{1,1,1},{1,1,2},{1,2,1},{1,2,2},
    {2,0,2},{2,1,1},{2,1,2},{2,2,0},{2,2,1},{2,2,2},
    // it2 (3 paths)
    {0,0,0},{1,1,0},{2,2,0}
};
__device__ __constant__ int L_OFF[3] = {0, 1, 4};

// ---------------------------------------------------------------------------
// On-device construction of the real-spherical-harmonic CG tensor (9x9x9).
// One launch of 729 threads; deterministic, runs in microseconds.
// ---------------------------------------------------------------------------
__device__ double dev_fact(int n) {
    double r = 1.0;
    for (int i = 2; i <= n; ++i) r *= (double)i;
    return r;
}

__device__ double dev_cg(int j1, int m1, int j2, int m2, int j3, int m3) {
    if (m1 + m2 != m3) return 0.0;
    int lo = (j1 > j2) ? (j1 - j2) : (j2 - j1);
    if (j3 < lo || j3 > j1 + j2) return 0.0;
    double pre = sqrt((2.0 * j3 + 1.0) * dev_fact(j1 + j2 - j3) * dev_fact(j1 - j2 + j3) *
                      dev_fact(-j1 + j2 + j3) / dev_fact(j1 + j2 + j3 + 1));
    pre *= sqrt(dev_fact(j3 + m3) * dev_fact(j3 - m3) * dev_fact(j1 - m1) *
                dev_fact(j1 + m1) * dev_fact(j2 - m2) * dev_fact(j2 + m2));
    double s = 0.0;
    for (int k = 0; k <= j1 + j2 + j3; ++k) {
        int d0 = k, d1 = j1 + j2 - j3 - k, d2 = j1 - m1 - k;
        int d3 = j2 + m2 - k, d4 = j3 - j2 + m1 + k, d5 = j3 - j1 - m2 + k;
        if (d0 < 0 || d1 < 0 || d2 < 0 || d3 < 0 || d4 < 0 || d5 < 0) continue;
        double term = 1.0 / (dev_fact(d0) * dev_fact(d1) * dev_fact(d2) *
                             dev_fact(d3) * dev_fact(d4) * dev_fact(d5));
        s += (k & 1) ? -term : term;
    }
    return pre * s;
}

// U[i][a] entry of the complex->real SH change-of-basis matrix for given l.
__device__ void u_entry(int l, int i, int a, double* re, double* im) {
    int m = i - l;
    const double s2 = 0.70710678118654752440;
    *re = 0.0; *im = 0.0;
    if (m == 0) {
        if (a == l) *re = 1.0;
    } else if (m > 0) {
        if (a == l + m)      *re = (m & 1) ? -s2 : s2;
        else if (a == l - m) *re = s2;
    } else {
        if (a == l + m)      *im = s2;
        else if (a == l - m) *im = -(((-m) & 1) ? -s2 : s2);
    }
}

__global__ void build_cgc_kernel(float* __restrict__ cgc) {
    int t = blockIdx.x * blockDim.x + threadIdx.x;
    if (t >= 729) return;
    int i = t / 81, j = (t / 9) % 9, k = t % 9;
    int l1 = (i >= 4) ? 2 : (i >= 1 ? 1 : 0);
    int l2 = (j >= 4) ? 2 : (j >= 1 ? 1 : 0);
    int l3 = (k >= 4) ? 2 : (k >= 1 ? 1 : 0);
    int o1 = (l1 == 2) ? 4 : l1, o2 = (l2 == 2) ? 4 : l2, o3 = (l3 == 2) ? 4 : l3;
    int li = i - o1, lj = j - o2, lk = k - o3;
    float out = 0.0f;
    int lo = (l1 > l2) ? (l1 - l2) : (l2 - l1);
    if (l3 >= lo && l3 <= l1 + l2) {
        double accre = 0.0, accim = 0.0;
        for (int a = 0; a < 2 * l1 + 1; ++a) {
            for (int b = 0; b < 2 * l2 + 1; ++b) {
                int m1 = a - l1, m2 = b - l2, m3 = m1 + m2;
                int c = m3 + l3;
                if (c < 0 || c > 2 * l3) continue;
                double g = dev_cg(l1, m1, l2, m2, l3, m3);
                if (g == 0.0) continue;
                double r1, i1, r2, i2, r3, i3;
                u_entry(l1, li, a, &r1, &i1);
                u_entry(l2, lj, b, &r2, &i2);
                u_entry(l3, lk, c, &r3, &i3);
                double tr = r1 * r2 - i1 * i2;
                double ti = r1 * i2 + i1 * r2;
                double fr = tr * r3 + ti * i3;   // * conj(u3)
                double fi = ti * r3 - tr * i3;
                accre += g * fr;
                accim += g * fi;
            }
        }
        out = (float)(accre + accim);  // matches reference blk.real + blk.imag
    }
    cgc[t] = out;
}

// ---------------------------------------------------------------------------
// Edge geometry: radial basis (with cutoff+mask folded in) and real SH.
// ---------------------------------------------------------------------------
__global__ __launch_bounds__(256) void edge_geom_kernel(
    const float* __restrict__ dr, const int* __restrict__ idx,
    float* __restrict__ rad, float* __restrict__ Yb, int E) {
    int e = blockIdx.x * blockDim.x + threadIdx.x;
    if (e >= E) return;
    float dx = dr[e * 3 + 0], dy = dr[e * 3 + 1], dz = dr[e * 3 + 2];
    float r = sqrtf(dx * dx + dy * dy + dz * dz + 1e-12f);
    float inv = 1.0f / r;
    float x = dx * inv, y = dy * inv, z = dz * inv;
    int ii = idx[e], jj = idx[E + e];
    float mask = (ii != jj) ? 1.0f : 0.0f;
    float t = fminf(r * (1.0f / RMAX), 1.0f - 1e-6f);
    float cut = (r < RMAX) ? __expf(1.0f - 1.0f / (1.0f - t * t)) : 0.0f;
    float scale = cut * mask * sqrtf(2.0f / RMAX) * inv;
    const float PI = 3.14159265358979f;
#pragma unroll
    for (int b = 0; b < NBASIS; ++b)
        rad[e * NBASIS + b] = scale * __sinf((float)(b + 1) * PI * r * (1.0f / RMAX));
    const float c0 = 0.28209479177387814f, c1 = 0.4886025119029199f;
    const float c2a = 1.0925484305920792f, c2b = 0.31539156525252005f;
    const float c2c = 0.5462742152960396f;
    float* Ye = Yb + e * 9;
    Ye[0] = c0;
    Ye[1] = c1 * y;
    Ye[2] = c1 * z;
    Ye[3] = c1 * x;
    Ye[4] = c2a * x * y;
    Ye[5] = c2a * y * z;
    Ye[6] = c2b * (3.0f * z * z - 1.0f);
    Ye[7] = c2a * x * z;
    Ye[8] = c2c * (x * x - y * y);
}

// ---------------------------------------------------------------------------
// Node feature init: channel 0 <- embed[Z[n]], channels 1..8 <- 0
// ---------------------------------------------------------------------------
__global__ void init_x_kernel(const float* __restrict__ embed, const int* __restrict__ Z,
                              float* __restrict__ xbuf, int N) {
    int t = blockIdx.x * blockDim.x + threadIdx.x;
    if (t >= N * 9 * FEAT) return;
    int f = t & 31;
    int nc = t >> 5;
    int ch = nc % 9;
    int n = nc / 9;
    xbuf[t] = (ch == 0) ? embed[Z[n] * FEAT + f] : 0.0f;
}

// ---------------------------------------------------------------------------
// Fused zero + eadd: message accumulator starts from the self contribution.
// ybuf[n,ch,:] = (ch < dAdd) ? xbuf[n,ch,:] : 0
// ---------------------------------------------------------------------------
__global__ void init_y_kernel(float* __restrict__ ybuf, const float* __restrict__ xbuf,
                              int N, int dAdd) {
    int t = blockIdx.x * blockDim.x + threadIdx.x;
    if (t >= N * 9 * FEAT) return;
    int ch = (t >> 5) % 9;
    ybuf[t] = (ch < dAdd) ? xbuf[t] : 0.0f;
}

// ---------------------------------------------------------------------------
// Edge message + scatter. One thread per (edge, feature). Sparse CG contraction
// in VALU fp32; scatter-add into node accumulator with HW fp32 atomics (L2).
// ---------------------------------------------------------------------------
__global__ __launch_bounds__(256) void msg_kernel(
    const float* __restrict__ x, const float* __restrict__ rad,
    const float* __restrict__ Yb, const float* __restrict__ cgc,
    const int* __restrict__ src, const int* __restrict__ dst,
    const float* __restrict__ W, int pbase, int np, int dIn, int dOut,
    float* __restrict__ ybuf, int E) {
    int t = blockIdx.x * blockDim.x + threadIdx.x;
    int e = t >> 5;
    int f = t & 31;
    if (e >= E) return;
    int s = src[e], d = dst[e];

    float radv[NBASIS];
#pragma unroll
    for (int b = 0; b < NBASIS; ++b) radv[b] = rad[e * NBASIS + b];
    float Yv[9];
#pragma unroll
    for (int b = 0; b < 9; ++b) Yv[b] = Yb[e * 9 + b];
    float xs[9];
#pragma unroll
    for (int a = 0; a < 9; ++a) xs[a] = (a < dIn) ? x[(s * 9 + a) * FEAT + f] : 0.0f;
    float msg[9];
#pragma unroll
    for (int c = 0; c < 9; ++c) msg[c] = 0.0f;

    for (int p = 0; p < np; ++p) {
        int l1 = PATHS_L[pbase + p][0];
        int l2 = PATHS_L[pbase + p][1];
        int l3 = PATHS_L[pbase + p][2];
        float w = 0.0f;
#pragma unroll
        for (int b = 0; b < NBASIS; ++b)
            w = fmaf(radv[b], W[(p * NBASIS + b) * FEAT + f], w);
        int o1 = L_OFF[l1], o2 = L_OFF[l2], o3 = L_OFF[l3];
        int n1 = 2 * l1 + 1, n2 = 2 * l2 + 1, n3 = 2 * l3 + 1;
        for (int a = 0; a < n1; ++a) {
            float xa = xs[o1 + a];
            for (int b = 0; b < n2; ++b) {
                float xyw = xa * Yv[o2 + b] * w;
                for (int c = 0; c < n3; ++c) {
                    float g = cgc[((o1 + a) * 9 + (o2 + b)) * 9 + (o3 + c)];
                    msg[o3 + c] = fmaf(g, xyw, msg[o3 + c]);
                }
            }
        }
    }
    for (int c = 0; c < dOut; ++c)
        unsafeAtomicAdd(&ybuf[(d * 9 + c) * FEAT + f], msg[c]);
}

// ---------------------------------------------------------------------------
// Dense 32x32 layer as a WMMA fp32 GEMM: Y[r, :] = X[r, :] @ K (+ bias on ch0).
// Weights staged in LDS via async-LDS DMA (when available), then hoisted into
// registers once per wave; grid-stride over 16-row tiles. Per tile: all 8 A
// fragments loaded up-front (clauseable global_load_b64 burst, single wait),
// then 16x V_WMMA_F32_16X16X4_F32 back-to-back.
// pitch = row stride in floats (32 for d=9 compact, 288 for d=1).
// ---------------------------------------------------------------------------
__global__ __launch_bounds__(256) void dense_wmma_kernel(
    const float* __restrict__ X, const float* __restrict__ Kw,
    const float* __restrict__ bias, float* __restrict__ Yo,
    int R, int pitch, int chMod) {
    __shared__ __align__(16) float ldsK[FEAT * FEAT];
#ifdef USE_ASYNC_LDS
    {
        // 256 threads x 16B = full 32x32 fp32 weight tile, via async LDS DMA.
        __builtin_amdgcn_global_load_async_to_lds_b128(
            (gv4i*)(Kw + threadIdx.x * 4),
            (lv4i*)(ldsK + threadIdx.x * 4),
            0, 0);
        __builtin_amdgcn_s_wait_asynccnt(0);
    }
#else
    for (int i = threadIdx.x; i < FEAT * FEAT; i += blockDim.x) ldsK[i] = Kw[i];
#endif
    __syncthreads();

    int wave = threadIdx.x >> 5;
    int lane = threadIdx.x & 31;
    int mrow = lane & 15;
    int khalf = lane >> 4;  // 0: K pair {0,1}; 1: K pair {2,3}

    // Hoist all B fragments (8 K-steps x 2 col tiles) into registers.
    v2f bf0[8], bf1[8];
#pragma unroll
    for (int ks = 0; ks < 8; ++ks) {
        int k0 = ks * 4;
        bf0[ks].x = ldsK[(k0 + 2 * khalf + 0) * FEAT + mrow];
        bf0[ks].y = ldsK[(k0 + 2 * khalf + 1) * FEAT + mrow];
        bf1[ks].x = ldsK[(k0 + 2 * khalf + 0) * FEAT + 16 + mrow];
        bf1[ks].y = ldsK[(k0 + 2 * khalf + 1) * FEAT + 16 + mrow];
    }

    int nTiles = R >> 4;  // R is a multiple of 16
    int stride = gridDim.x * (blockDim.x >> 5);
    for (int tile = blockIdx.x * (blockDim.x >> 5) + wave; tile < nTiles; tile += stride) {
        int rbase = tile * 16;
        const float* xr = X + (size_t)(rbase + mrow) * pitch + 2 * khalf;

        // Prefetch the next tile's A rows into cache (global_prefetch_b8).
        int nxt = tile + stride;
        if (nxt < nTiles)
            __builtin_prefetch(X + (size_t)(nxt * 16 + mrow) * pitch, 0, 3);

        // Load all 8 A fragments first: one burst of global_load_b64, one wait.
        v2f af[8];
#pragma unroll
        for (int ks = 0; ks < 8; ++ks) {
            af[ks].x = xr[ks * 4 + 0];
            af[ks].y = xr[ks * 4 + 1];
        }

        v8f acc0 = {0.f, 0.f, 0.f, 0.f, 0.f, 0.f, 0.f, 0.f};
        v8f acc1 = {0.f, 0.f, 0.f, 0.f, 0.f, 0.f, 0.f, 0.f};
#pragma unroll
        for (int ks = 0; ks < 8; ++ks) {
            acc0 = __builtin_amdgcn_wmma_f32_16x16x4_f32(false, af[ks], false, bf0[ks],
                                                         (short)0, acc0, false, false);
            acc1 = __builtin_amdgcn_wmma_f32_16x16x4_f32(false, af[ks], false, bf1[ks],
                                                         (short)0, acc1, false, false);
        }

#pragma unroll
        for (int v = 0; v < 8; ++v) {
            int row = rbase + v + 8 * khalf;
            float d0 = acc0[v];
            float d1 = acc1[v];
            if (row % chMod == 0) {  // bias only on angular channel 0
                d0 += bias[mrow];
                d1 += bias[16 + mrow];
            }
            Yo[(size_t)row * pitch + mrow] = d0;
            Yo[(size_t)row * pitch + 16 + mrow] = d1;
        }
    }
}

// ---------------------------------------------------------------------------
// Gated SiLU: ch0 -> silu(s); ch>0 -> v * sigmoid(s), s = ch0 pre-activation.
// ---------------------------------------------------------------------------
__global__ void silu_kernel(const float* __restrict__ t1, float* __restrict__ t2,
                            int N, int d) {
    int t = blockIdx.x * blockDim.x + threadIdx.x;
    if (t >= N * d * FEAT) return;
    int f = t & 31;
    int nc = t >> 5;
    int ch = nc % d;
    int n = nc / d;
    float s = t1[(n * 9 + 0) * FEAT + f];
    float sig = 1.0f / (1.0f + __expf(-s));
    int idx = (n * 9 + ch) * FEAT + f;
    float v = t1[idx];
    t2[idx] = (ch == 0) ? s * sig : v * sig;
}

// ---------------------------------------------------------------------------
// Residual: xbuf[n, ch, :] = (ch < dOld ? xbuf : 0) + t3[n, ch, :], ch < dNew
// ---------------------------------------------------------------------------
__global__ void residual_kernel(float* __restrict__ xbuf, const float* __restrict__ t3,
                                int N, int dNew, int dOld) {
    int t = blockIdx.x * blockDim.x + threadIdx.x;
    if (t >= N * dNew * FEAT) return;
    int f = t & 31;
    int nc = t >> 5;
    int ch = nc % dNew;
    int n = nc / dNew;
    int idx = (n * 9 + ch) * FEAT + f;
    float prev = (ch < dOld) ? xbuf[idx] : 0.0f;
    xbuf[idx] = prev + t3[idx];
}

__global__ void out_kernel(const float* __restrict__ xbuf, float* __restrict__ out, int N) {
    int t = blockIdx.x * blockDim.x + threadIdx.x;
    if (t >= N * FEAT) return;
    int n = t >> 5, f = t & 31;
    out[t] = xbuf[n * 9 * FEAT + f];
}

// ---------------------------------------------------------------------------
// Host launcher
// ---------------------------------------------------------------------------
extern "C" void kernel_launch(void* const* d_in, const int* in_sizes, int n_in,
                              void* d_out, int out_size, void* d_ws, size_t ws_size,
                              hipStream_t stream) {
    const float* dr_vec = (const float*)d_in[0];
    const int* Z = (const int*)d_in[1];
    const int* nbr = (const int*)d_in[2];  // [0:E]=idx_i (dst), [E:2E]=idx_j (src)
    const float* embed = (const float*)d_in[3];
    const float* wmp[3] = {(const float*)d_in[4], (const float*)d_in[5], (const float*)d_in[6]};
    const float* dk1 = (const float*)d_in[7];
    const float* db1 = (const float*)d_in[8];
    const float* dk2 = (const float*)d_in[9];
    const float* db2 = (const float*)d_in[10];
    float* out = (float*)d_out;

    const int E = N_EDGES, N = N_NODES;
    const int* idx_i = nbr;      // dst
    const int* idx_j = nbr + E;  // src

    // Workspace layout (floats)
    float* ws = (float*)d_ws;
    float* cgc = ws;                       // 729 (pad to 768)
    float* rad = cgc + 768;                // E*8
    float* Yb = rad + (size_t)E * 8;       // E*9
    float* xbuf = Yb + (size_t)E * 9;      // N*9*32
    float* ybuf = xbuf + (size_t)N * 288;  // N*9*32
    float* t1 = ybuf + (size_t)N * 288;    // N*9*32
    float* t2 = t1 + (size_t)N * 288;      // N*9*32
    float* t3 = t2 + (size_t)N * 288;      // N*9*32

    // 1) CG tensor
    build_cgc_kernel<<<3, 256, 0, stream>>>(cgc);
    // 2) Edge geometry
    edge_geom_kernel<<<(E + 255) / 256, 256, 0, stream>>>(dr_vec, nbr, rad, Yb, E);
    // 3) Node init
    init_x_kernel<<<(N * 288 + 255) / 256, 256, 0, stream>>>(embed, Z, xbuf, N);

    // Per-iteration parameters
    const int pbase[3] = {0, 3, 18};
    const int np[3] = {3, 15, 3};
    const int dIn[3] = {1, 9, 9};
    const int dOut[3] = {9, 9, 1};
    const int dDense[3] = {9, 9, 1};
    const int dAdd[3] = {1, 9, 1};
    const int dOld[3] = {1, 9, 1};

    for (int it = 0; it < 3; ++it) {
        int R = N * dDense[it];
        int pitch = (dDense[it] == 9) ? FEAT : 9 * FEAT;
        int chMod = (dDense[it] == 9) ? 9 : 1;
        int nTiles = R / 16;
        int dBlocks = (nTiles + 15) / 16;  // ~2 tiles per wave (8 waves/block)

        // a) y <- eadd base (self contribution; rest zero)
        init_y_kernel<<<(N * 288 + 255) / 256, 256, 0, stream>>>(ybuf, xbuf, N, dAdd[it]);
        // b) edge messages + scatter
        msg_kernel<<<(E * 32 + 255) / 256, 256, 0, stream>>>(
            xbuf, rad, Yb, cgc, idx_j, idx_i, wmp[it],
            pbase[it], np[it], dIn[it], dOut[it], ybuf, E);
        // c) t1 = y @ dk1 + db1(ch0)
        dense_wmma_kernel<<<dBlocks, 256, 0, stream>>>(
            ybuf, dk1 + it * FEAT * FEAT, db1 + it * FEAT, t1, R, pitch, chMod);
        // d) t2 = gated silu(t1)
        silu_kernel<<<(N * dDense[it] * FEAT + 255) / 256, 256, 0, stream>>>(
            t1, t2, N, dDense[it]);
        // e) t3 = t2 @ dk2 + db2(ch0)
        dense_wmma_kernel<<<dBlocks, 256, 0, stream>>>(
            t2, dk2 + it * FEAT * FEAT, db2 + it * FEAT, t3, R, pitch, chMod);
        // f) x = eadd(x, t3)
        residual_kernel<<<(N * dDense[it] * FEAT + 255) / 256, 256, 0, stream>>>(
            xbuf, t3, N, dDense[it], dOld[it]);
    }

    // 4) Output
    out_kernel<<<(N * FEAT + 255) / 256, 256, 0, stream>>>(xbuf, out, N);
}